// GCN_63204738728336
// MI455X (gfx1250) — compile-verified
//
#include <hip/hip_runtime.h>
#include <hip/hip_bf16.h>

typedef float v2f __attribute__((ext_vector_type(2)));
typedef float v8f __attribute__((ext_vector_type(8)));

#define LN_EPS    1e-5f
#define NEG_SLOPE 0.2f

// ---- ordered-uint encoding for float atomicMax (all finite values) ----
__device__ __forceinline__ unsigned ford(float f) {
    unsigned u = __float_as_uint(f);
    return (u & 0x80000000u) ? ~u : (u | 0x80000000u);
}
__device__ __forceinline__ float uord(unsigned u) {
    return (u & 0x80000000u) ? __uint_as_float(u & 0x7FFFFFFFu) : __uint_as_float(~u);
}

// ---------------------------------------------------------------------
// init: zero accumulators / out, deg=1 (self loop), m=-inf(encoded 0)
// ---------------------------------------------------------------------
__global__ __launch_bounds__(256) void k_init(float* __restrict__ agg,
                                              float* __restrict__ out,
                                              float* __restrict__ deg,
                                              unsigned* __restrict__ mbuf,
                                              float* __restrict__ denom, int N) {
    int i = blockIdx.x * 256 + threadIdx.x;
    int tot = N * 64;
    if (i < tot) { agg[i] = 0.f; out[i] = 0.f; }
    if (i < N)   { deg[i] = 1.0f; mbuf[i] = 0u; denom[i] = 0.f; }
}

// ---------------------------------------------------------------------
// degree count over dst (edge list streamed NT; counters L2-resident)
// ---------------------------------------------------------------------
__global__ __launch_bounds__(256) void k_deg(const int* __restrict__ ei,
                                             float* __restrict__ deg, int E) {
    int i = blockIdx.x * 256 + threadIdx.x;
    if (i < E) {
        int d = __builtin_nontemporal_load(ei + E + i);
        atomicAdd(deg + d, 1.0f);
    }
}

__global__ __launch_bounds__(256) void k_dinv(float* __restrict__ deg, int N) {
    int i = blockIdx.x * 256 + threadIdx.x;
    if (i < N) { float g = deg[i]; deg[i] = (g > 0.f) ? rsqrtf(g) : 0.f; }
}

// ---------------------------------------------------------------------
// LDS staging permutation: W[4kk+2h+j][c] -> sWp[((kk*2+h)*64 + c)*2 + j]
// so each WMMA B fragment is one aligned ds_load_b64.
// ---------------------------------------------------------------------
__device__ __forceinline__ void stage_w(float* __restrict__ sWp,
                                        const float* __restrict__ W) {
    for (int i = threadIdx.x; i < 4096; i += 256) {
        int k = i >> 6, c = i & 63;
        int kk = k >> 2, h = (k >> 1) & 1, j = k & 1;
        sWp[((((kk << 1) + h) << 6) + c) * 2 + j] = W[i];
    }
}

// ---------------------------------------------------------------------
// C[N,64] = A[N,64] @ W[64,64]  via V_WMMA_F32_16X16X4_F32, fp32 exact.
// One wave = 16-row strip, 4 N-tiles, K chained 16x (64 WMMA / strip).
// ---------------------------------------------------------------------
__global__ __launch_bounds__(256) void k_gemm64(const float* __restrict__ A,
                                                const float* __restrict__ W,
                                                float* __restrict__ C, int nstrips) {
    __shared__ float sWp[64 * 64];
    stage_w(sWp, W);
    __syncthreads();

    int lane  = threadIdx.x & 31;
    int strip = blockIdx.x * 8 + (threadIdx.x >> 5);
    if (strip >= nstrips) return;
    int r = lane & 15, h = lane >> 4;

    const float* ap = A + (size_t)(strip * 16 + r) * 64 + 2 * h;
    v8f zero = {0.f, 0.f, 0.f, 0.f, 0.f, 0.f, 0.f, 0.f};
    v8f acc[4] = {zero, zero, zero, zero};

    for (int kk = 0; kk < 16; ++kk) {
        v2f a = *(const v2f*)(ap + kk * 4);  // A[row][4kk+2h], [4kk+2h+1]
        const v2f* bp = (const v2f*)&sWp[((((kk << 1) + h) << 6) + r) << 1];
#pragma unroll
        for (int t = 0; t < 4; ++t) {
            acc[t] = __builtin_amdgcn_wmma_f32_16x16x4_f32(
                false, a, false, bp[t * 16], (short)0, acc[t], false, false);
        }
    }
    float* cp = C + (size_t)strip * 16 * 64;
#pragma unroll
    for (int t = 0; t < 4; ++t)
#pragma unroll
        for (int g = 0; g < 8; ++g)
            cp[(size_t)(g + 8 * h) * 64 + t * 16 + r] = acc[t][g];
}

// Fused dual GEMM for GATv2: Cl = A@Wl, Cr = A@Wr (shared A loads).
__global__ __launch_bounds__(256) void k_gemm64_dual(const float* __restrict__ A,
                                                     const float* __restrict__ Wl,
                                                     const float* __restrict__ Wr,
                                                     float* __restrict__ Cl,
                                                     float* __restrict__ Cr, int nstrips) {
    __shared__ float sWp[2 * 64 * 64];
    stage_w(sWp, Wl);
    stage_w(sWp + 4096, Wr);
    __syncthreads();

    int lane  = threadIdx.x & 31;
    int strip = blockIdx.x * 8 + (threadIdx.x >> 5);
    if (strip >= nstrips) return;
    int r = lane & 15, h = lane >> 4;

    const float* ap = A + (size_t)(strip * 16 + r) * 64 + 2 * h;
    v8f zero = {0.f, 0.f, 0.f, 0.f, 0.f, 0.f, 0.f, 0.f};
    v8f al[4] = {zero, zero, zero, zero};
    v8f ar[4] = {zero, zero, zero, zero};

    for (int kk = 0; kk < 16; ++kk) {
        v2f a = *(const v2f*)(ap + kk * 4);
        const v2f* bpl = (const v2f*)&sWp[((((kk << 1) + h) << 6) + r) << 1];
        const v2f* bpr = bpl + 2048;  // +4096 floats
#pragma unroll
        for (int t = 0; t < 4; ++t) {
            al[t] = __builtin_amdgcn_wmma_f32_16x16x4_f32(
                false, a, false, bpl[t * 16], (short)0, al[t], false, false);
            ar[t] = __builtin_amdgcn_wmma_f32_16x16x4_f32(
                false, a, false, bpr[t * 16], (short)0, ar[t], false, false);
        }
    }
    float* cl = Cl + (size_t)strip * 16 * 64;
    float* cr = Cr + (size_t)strip * 16 * 64;
#pragma unroll
    for (int t = 0; t < 4; ++t)
#pragma unroll
        for (int g = 0; g < 8; ++g) {
            cl[(size_t)(g + 8 * h) * 64 + t * 16 + r] = al[t][g];
            cr[(size_t)(g + 8 * h) * 64 + t * 16 + r] = ar[t][g];
        }
}

// scalar tail for rows not covered by full 16-row strips (N%16, here 0)
__global__ __launch_bounds__(256) void k_gemm_tail(const float* __restrict__ A,
                                                   const float* __restrict__ W,
                                                   float* __restrict__ C,
                                                   int row0, int nrows) {
    int i = blockIdx.x * 256 + threadIdx.x;
    int rows = nrows - row0;
    if (i >= rows * 64) return;
    int rr = row0 + i / 64, c = i & 63;
    float s = 0.f;
    for (int k = 0; k < 64; ++k) s += A[(size_t)rr * 64 + k] * W[k * 64 + c];
    C[(size_t)rr * 64 + c] = s;
}

// ---------------------------------------------------------------------
// GCN aggregate: agg[dst] += h[src]*dinv[src]*dinv[dst]   (16 thr/edge)
// edge ids E..E+N-1 are the implicit self loops
// ---------------------------------------------------------------------
__global__ __launch_bounds__(256) void k_gcn_agg(const int* __restrict__ ei,
                                                 const float* __restrict__ h,
                                                 const float* __restrict__ dinv,
                                                 float* __restrict__ agg, int E, int N) {
    int gid = blockIdx.x * 256 + threadIdx.x;
    int eid = gid >> 4, sub = gid & 15;
    if (eid >= E + N) return;
    int s, d;
    if (eid < E) {
        s = __builtin_nontemporal_load(ei + eid);
        d = __builtin_nontemporal_load(ei + E + eid);
    } else {
        s = d = eid - E;
    }
    float norm = dinv[s] * dinv[d];
    float4 v = *(const float4*)(h + (size_t)s * 64 + sub * 4);  // L2-resident gather
    float* ap = agg + (size_t)d * 64 + sub * 4;
    atomicAdd(ap + 0, v.x * norm);
    atomicAdd(ap + 1, v.y * norm);
    atomicAdd(ap + 2, v.z * norm);
    atomicAdd(ap + 3, v.w * norm);
}

// ---------------------------------------------------------------------
// per-node: relu(agg + b1) -> LayerNorm -> hout   (wave32 per node)
// ---------------------------------------------------------------------
__global__ __launch_bounds__(256) void k_node_ln(const float* __restrict__ agg,
                                                 const float* __restrict__ b1,
                                                 const float* __restrict__ lnw,
                                                 const float* __restrict__ lnb,
                                                 float* __restrict__ hout, int N) {
    int lane = threadIdx.x & 31;
    int n = blockIdx.x * 8 + (threadIdx.x >> 5);
    if (n >= N) return;
    const float* p = agg + (size_t)n * 64 + lane * 2;
    float vx = fmaxf(p[0] + b1[lane * 2], 0.f);
    float vy = fmaxf(p[1] + b1[lane * 2 + 1], 0.f);
    float s = vx + vy;
    for (int m = 16; m >= 1; m >>= 1) s += __shfl_xor(s, m, 32);
    float mu = s * (1.f / 64.f);
    float dx = vx - mu, dy = vy - mu;
    float q = dx * dx + dy * dy;
    for (int m = 16; m >= 1; m >>= 1) q += __shfl_xor(q, m, 32);
    float inv = rsqrtf(q * (1.f / 64.f) + LN_EPS);
    float* o = hout + (size_t)n * 64 + lane * 2;
    o[0] = dx * inv * lnw[lane * 2] + lnb[lane * 2];
    o[1] = dy * inv * lnw[lane * 2 + 1] + lnb[lane * 2 + 1];
}

// ---------------------------------------------------------------------
// GATv2 score: e = att . leaky_relu(xl[src] + xr[dst]); seg-max into mbuf
// 16 threads/edge, half-wave shuffle reduction
// ---------------------------------------------------------------------
__global__ __launch_bounds__(256) void k_att_score(const int* __restrict__ ei,
                                                   const float* __restrict__ xl,
                                                   const float* __restrict__ xr,
                                                   const float* __restrict__ att,
                                                   float* __restrict__ ebuf,
                                                   unsigned* __restrict__ mbuf,
                                                   int E, int N) {
    int gid = blockIdx.x * 256 + threadIdx.x;
    int eid = gid >> 4, sub = gid & 15;
    if (eid >= E + N) return;
    int s, d;
    if (eid < E) {
        s = __builtin_nontemporal_load(ei + eid);
        d = __builtin_nontemporal_load(ei + E + eid);
    } else {
        s = d = eid - E;
    }
    float4 a = *(const float4*)(xl + (size_t)s * 64 + sub * 4);
    float4 b = *(const float4*)(xr + (size_t)d * 64 + sub * 4);
    float4 w = *(const float4*)(att + sub * 4);
    float p = 0.f, v;
    v = a.x + b.x; v = v > 0.f ? v : NEG_SLOPE * v; p += w.x * v;
    v = a.y + b.y; v = v > 0.f ? v : NEG_SLOPE * v; p += w.y * v;
    v = a.z + b.z; v = v > 0.f ? v : NEG_SLOPE * v; p += w.z * v;
    v = a.w + b.w; v = v > 0.f ? v : NEG_SLOPE * v; p += w.w * v;
    for (int m = 8; m >= 1; m >>= 1) p += __shfl_xor(p, m, 16);
    if (sub == 0) {
        __builtin_nontemporal_store(p, ebuf + eid);
        atomicMax(mbuf + d, ford(p));
    }
}

// softmax numerator + denominator accumulation (1 thread/edge)
__global__ __launch_bounds__(256) void k_att_norm(const int* __restrict__ ei,
                                                  float* __restrict__ ebuf,
                                                  const unsigned* __restrict__ mbuf,
                                                  float* __restrict__ denom,
                                                  int E, int N) {
    int i = blockIdx.x * 256 + threadIdx.x;
    if (i >= E + N) return;
    int d = (i < E) ? __builtin_nontemporal_load(ei + E + i) : (i - E);
    float m = uord(mbuf[d]);
    float ex = __expf(ebuf[i] - m);
    ebuf[i] = ex;
    atomicAdd(denom + d, ex);
}

// out[dst] += alpha * xl[src]   (16 thr/edge)
__global__ __launch_bounds__(256) void k_att_agg(const int* __restrict__ ei,
                                                 const float* __restrict__ ebuf,
                                                 const float* __restrict__ denom,
                                                 const float* __restrict__ xl,
                                                 float* __restrict__ out, int E, int N) {
    int gid = blockIdx.x * 256 + threadIdx.x;
    int eid = gid >> 4, sub = gid & 15;
    if (eid >= E + N) return;
    int s, d;
    if (eid < E) {
        s = __builtin_nontemporal_load(ei + eid);
        d = __builtin_nontemporal_load(ei + E + eid);
    } else {
        s = d = eid - E;
    }
    float alpha = ebuf[eid] / denom[d];
    float4 v = *(const float4*)(xl + (size_t)s * 64 + sub * 4);
    float* op = out + (size_t)d * 64 + sub * 4;
    atomicAdd(op + 0, alpha * v.x);
    atomicAdd(op + 1, alpha * v.y);
    atomicAdd(op + 2, alpha * v.z);
    atomicAdd(op + 3, alpha * v.w);
}

__global__ __launch_bounds__(256) void k_finalize(float* __restrict__ out,
                                                  const float* __restrict__ b2, int tot) {
    int i = blockIdx.x * 256 + threadIdx.x;
    if (i < tot) out[i] = fmaxf(out[i] + b2[i & 63], 0.f);
}

// ---------------------------------------------------------------------
extern "C" void kernel_launch(void* const* d_in, const int* in_sizes, int n_in,
                              void* d_out, int out_size, void* d_ws, size_t ws_size,
                              hipStream_t stream) {
    const float* x   = (const float*)d_in[0];
    const int*   ei  = (const int*)d_in[1];   // [2,E]: row0=src, row1=dst
    const float* W1  = (const float*)d_in[2];
    const float* b1  = (const float*)d_in[3];
    const float* lnw = (const float*)d_in[4];
    const float* lnb = (const float*)d_in[5];
    const float* Wl  = (const float*)d_in[6];
    const float* Wr  = (const float*)d_in[7];
    const float* att = (const float*)d_in[8];
    const float* b2  = (const float*)d_in[9];
    float* out = (float*)d_out;

    const int N = in_sizes[0] / 64;
    const int E = in_sizes[1] / 2;
    const int EN = E + N;

    // workspace carve-out (~85 MB): node-feature arrays fit in 192 MB L2
    char* ws = (char*)d_ws;
    size_t off = 0;
    auto carve = [&](size_t bytes) -> void* {
        void* p = ws + off;
        off += (bytes + 255) & ~(size_t)255;
        return p;
    };
    float*    hbuf  = (float*)carve((size_t)N * 64 * 4);  // h (GCN out, LN out)
    float*    aggxl = (float*)carve((size_t)N * 64 * 4);  // GCN agg, reused as xl
    float*    xrb   = (float*)carve((size_t)N * 64 * 4);  // xr
    float*    ebuf  = (float*)carve((size_t)EN * 4);      // e / exp(e-m)
    unsigned* mbuf  = (unsigned*)carve((size_t)N * 4);    // seg-max (ordered uint)
    float*    denom = (float*)carve((size_t)N * 4);
    float*    degv  = (float*)carve((size_t)N * 4);       // deg -> dinv in place

    const int tot = N * 64;
    const int egrid = (int)(((long long)EN * 16 + 255) / 256);
    const int nstrips = N / 16, rem = N - nstrips * 16;

    k_init<<<(tot + 255) / 256, 256, 0, stream>>>(aggxl, out, degv, mbuf, denom, N);
    k_deg<<<(E + 255) / 256, 256, 0, stream>>>(ei, degv, E);
    k_dinv<<<(N + 255) / 256, 256, 0, stream>>>(degv, N);

    k_gemm64<<<(nstrips + 7) / 8, 256, 0, stream>>>(x, W1, hbuf, nstrips);
    if (rem) k_gemm_tail<<<(rem * 64 + 255) / 256, 256, 0, stream>>>(x, W1, hbuf, nstrips * 16, N);

    k_gcn_agg<<<egrid, 256, 0, stream>>>(ei, hbuf, degv, aggxl, E, N);
    k_node_ln<<<(N + 7) / 8, 256, 0, stream>>>(aggxl, b1, lnw, lnb, hbuf, N);

    k_gemm64_dual<<<(nstrips + 7) / 8, 256, 0, stream>>>(hbuf, Wl, Wr, aggxl, xrb, nstrips);
    if (rem) {
        k_gemm_tail<<<(rem * 64 + 255) / 256, 256, 0, stream>>>(hbuf, Wl, aggxl, nstrips * 16, N);
        k_gemm_tail<<<(rem * 64 + 255) / 256, 256, 0, stream>>>(hbuf, Wr, xrb, nstrips * 16, N);
    }

    k_att_score<<<egrid, 256, 0, stream>>>(ei, aggxl, xrb, att, ebuf, mbuf, E, N);
    k_att_norm<<<(EN + 255) / 256, 256, 0, stream>>>(ei, ebuf, mbuf, denom, E, N);
    k_att_agg<<<egrid, 256, 0, stream>>>(ei, ebuf, denom, aggxl, out, E, N);
    k_finalize<<<(tot + 255) / 256, 256, 0, stream>>>(out, b2, tot);
}